// LTCCell_54820962566392
// MI455X (gfx1250) — compile-verified
//
#include <hip/hip_runtime.h>
#include <stdint.h>

#define UNITS   512
#define SENS    128
#define BATCH   128
#define MOTOR   64
#define UNFOLDS 6
#define EPS_F   1e-8f
#define TB      2           // batch rows per thread
#define CI      4           // i-rows staged per LDS chunk
#define NBUF    3           // LDS ring depth (issue distance 2)
#define NCHUNK  (UNITS / CI)

static_assert(CI == 4, "asm wait immediates assume CI==4");

// ---------- CDNA5 helpers ----------

#if __has_builtin(__builtin_amdgcn_tanhf)
#pragma message("CDNA5: using v_tanh_f32 builtin for sigmoid")
#else
#pragma message("CDNA5: tanh builtin NOT available, falling back to exp-based sigmoid")
#endif

// sigmoid via gfx1250 v_tanh_f32 when available (1 transcendental instead of 2)
__device__ __forceinline__ float fsigmoid(float t) {
#if __has_builtin(__builtin_amdgcn_tanhf)
    return fmaf(__builtin_amdgcn_tanhf(0.5f * t), 0.5f, 0.5f);
#else
    return 1.0f / (1.0f + __expf(-t));
#endif
}

// wave-wide async copy of 32 lanes x 16B global -> LDS (tracked by ASYNCcnt)
__device__ __forceinline__ void async_ld16(const void* g, void* l) {
    unsigned lds_off = (unsigned)(uintptr_t)l;   // low 32 bits of generic = LDS byte offset
    unsigned long long ga = (unsigned long long)(uintptr_t)g;
    asm volatile("global_load_async_to_lds_b128 %0, %1, off"
                 :: "v"(lds_off), "v"(ga) : "memory");
}

// ASYNCcnt waits: mid-loop 2 chunks (8 ops) in flight; drain points 4 / 0.
__device__ __forceinline__ void wait_async8() {
#if __has_builtin(__builtin_amdgcn_s_wait_asynccnt)
    __builtin_amdgcn_s_wait_asynccnt(2 * CI);
#else
    asm volatile("s_wait_asynccnt 8" ::: "memory");
#endif
}
__device__ __forceinline__ void wait_async4() {
#if __has_builtin(__builtin_amdgcn_s_wait_asynccnt)
    __builtin_amdgcn_s_wait_asynccnt(CI);
#else
    asm volatile("s_wait_asynccnt 4" ::: "memory");
#endif
}
__device__ __forceinline__ void wait_async0() {
#if __has_builtin(__builtin_amdgcn_s_wait_asynccnt)
    __builtin_amdgcn_s_wait_asynccnt(0);
#else
    asm volatile("s_wait_asynccnt 0" ::: "memory");
#endif
}

__device__ __forceinline__ void wait_ds0() {
#if __has_builtin(__builtin_amdgcn_s_wait_dscnt)
    __builtin_amdgcn_s_wait_dscnt(0);
#else
    asm volatile("s_wait_dscnt 0" ::: "memory");
#endif
}

// ---------- kernels ----------

// One ODE unfold. Block = 256 threads (one u each), TB batch rows.
// Parameter tiles stream through a 3-deep LDS ring via wave-private async
// global->LDS copies (each wave stages exactly the 32 columns its own lanes
// consume), issue distance 2, so the hot loop needs NO workgroup barriers —
// only the wave's own s_wait_asynccnt / s_wait_dscnt.
__global__ __launch_bounds__(256) void ltc_unfold(
    const float* __restrict__ src, float* __restrict__ dst,
    const float4* __restrict__ packedR,
    const float* __restrict__ gleak, const float* __restrict__ vleak,
    const float* __restrict__ cm,
    const float* __restrict__ snum, const float* __restrict__ sden) {
    __shared__ float4 pbuf[NBUF][CI * 256];   // 3 x 16 KB param ring
    __shared__ float  vbuf[TB][UNITS];        // 4 KB presynaptic state (read-only after 1 barrier)

    const int tid = threadIdx.x;
    const int u0  = blockIdx.x * 256;
    const int b0  = blockIdx.y * TB;
    const int u   = u0 + tid;

    // per-thread streaming pointer into the packed parameter table
    const float4* gp = packedR + u0 + tid;        // column (u), row 0

    // prologue: async-stage chunks 0 and 1, ASAP
#pragma unroll
    for (int c = 0; c < 2; ++c) {
#pragma unroll
        for (int r = 0; r < CI; ++r)
            async_ld16(gp + (c * CI + r) * UNITS, &pbuf[c][r * 256 + tid]);
    }
    gp += 2 * CI * UNITS;                         // now points at chunk 2

    // stage v_pre rows (TB contiguous rows = one coalesced float4 per thread)
    {
        const float4* s4 = (const float4*)(src + (size_t)b0 * UNITS);
        ((float4*)vbuf)[tid] = s4[tid];
    }
    __syncthreads();   // the only workgroup barrier in this kernel

    float num[TB], den[TB];
#pragma unroll
    for (int b = 0; b < TB; ++b) { num[b] = 0.f; den[b] = 0.f; }

    int bi = 0;                 // ring slot holding chunk j
    int ei = 2 % NBUF;          // ring slot receiving chunk j+2
    for (int j = 0; j < NCHUNK; ++j) {
        wait_ds0();             // prior LDS reads of the slot being refilled are done
        if (j + 2 < NCHUNK) {
            float4* nb = pbuf[ei];
#pragma unroll
            for (int r = 0; r < CI; ++r)
                async_ld16(gp + r * UNITS, &nb[r * 256 + tid]);
            gp += CI * UNITS;
            wait_async8();      // chunks j+1, j+2 may remain in flight; chunk j landed
        } else if (j + 1 < NCHUNK) {
            wait_async4();
        } else {
            wait_async0();
        }

        const float4* pb = pbuf[bi];
        const int i0 = j * CI;
#pragma unroll
        for (int r = 0; r < CI; ++r) {
            float4 p = pb[r * 256 + tid];       // ds_load_b128, wave-private columns
#pragma unroll
            for (int b = 0; b < TB; ++b) {
                float sg = fsigmoid(fmaf(vbuf[b][i0 + r], p.x, -p.y));
                num[b] = fmaf(p.z, sg, num[b]);
                den[b] = fmaf(p.w, sg, den[b]);
            }
        }
        bi = (bi + 1 == NBUF) ? 0 : bi + 1;
        ei = (ei + 1 == NBUF) ? 0 : ei + 1;
    }

    const float gl = gleak[u], vl = vleak[u], cmt = cm[u] * (float)UNFOLDS;
#pragma unroll
    for (int b = 0; b < TB; ++b) {
        int gi = (b0 + b) * UNITS + u;
        float wn = num[b] + snum[gi];
        float wd = den[b] + sden[gi];
        float vpre = vbuf[b][u];
        float numer = cmt * vpre + gl * vl + wn;
        float denom = cmt + gl + wd + EPS_F;
        dst[gi] = numer / denom;
    }
}

// Fuse {sigma, mu, w, erev, mask} -> float4 {a = sigma, c = mu*sigma, ws = w*mask*erev, wp = w*mask}
__global__ __launch_bounds__(256) void ltc_pack(
    const float* __restrict__ sg, const float* __restrict__ mu,
    const float* __restrict__ w,  const float* __restrict__ er,
    const float* __restrict__ mk, float4* __restrict__ out, int n) {
    int i = blockIdx.x * 256 + threadIdx.x;
    if (i < n) {
        float a  = sg[i];
        float c  = mu[i] * a;
        float wm = w[i] * mk[i];
        out[i] = make_float4(a, c, wm * er[i], wm);
    }
}

// Sensory reduction: snum/sden[b,u] = sum_s f(x[b,s]; packedS[s,u])
__global__ __launch_bounds__(256) void ltc_sensory(
    const float* __restrict__ inputs, const float* __restrict__ in_w,
    const float* __restrict__ in_b,   const float4* __restrict__ packedS,
    float* __restrict__ snum, float* __restrict__ sden) {
    __shared__ float xrow[SENS];
    const int tid = threadIdx.x;
    const int b = blockIdx.y;
    const int u = blockIdx.x * 256 + tid;
    if (tid < SENS) xrow[tid] = fmaf(inputs[b * SENS + tid], in_w[tid], in_b[tid]);
    __syncthreads();
    float num = 0.f, den = 0.f;
    for (int s = 0; s < SENS; ++s) {
        float4 p = packedS[s * UNITS + u];
        float sg = fsigmoid(fmaf(xrow[s], p.x, -p.y));
        num = fmaf(p.z, sg, num);
        den = fmaf(p.w, sg, den);
    }
    snum[b * UNITS + u] = num;
    sden[b * UNITS + u] = den;
}

// outputs = state[:, :MOTOR] * output_w + output_b
__global__ __launch_bounds__(256) void ltc_output(
    const float* __restrict__ state, const float* __restrict__ ow,
    const float* __restrict__ ob, float* __restrict__ out) {
    int idx = blockIdx.x * 256 + threadIdx.x;   // 8192
    int b = idx >> 6, u = idx & (MOTOR - 1);
    out[idx] = fmaf(state[b * UNITS + u], ow[u], ob[u]);
}

// ---------- launch ----------

extern "C" void kernel_launch(void* const* d_in, const int* in_sizes, int n_in,
                              void* d_out, int out_size, void* d_ws, size_t ws_size,
                              hipStream_t stream) {
    (void)in_sizes; (void)n_in; (void)out_size; (void)ws_size;
    const float* inputs  = (const float*)d_in[0];
    const float* state   = (const float*)d_in[1];
    const float* gleak   = (const float*)d_in[2];
    const float* vleak   = (const float*)d_in[3];
    const float* cm      = (const float*)d_in[4];
    const float* sigma   = (const float*)d_in[5];
    const float* mu      = (const float*)d_in[6];
    const float* w       = (const float*)d_in[7];
    const float* erev    = (const float*)d_in[8];
    const float* s_sigma = (const float*)d_in[9];
    const float* s_mu    = (const float*)d_in[10];
    const float* s_w     = (const float*)d_in[11];
    const float* s_erev  = (const float*)d_in[12];
    const float* in_w    = (const float*)d_in[13];
    const float* in_b    = (const float*)d_in[14];
    const float* out_w   = (const float*)d_in[15];
    const float* out_b   = (const float*)d_in[16];
    const float* mask    = (const float*)d_in[17];
    const float* s_mask  = (const float*)d_in[18];

    char* ws = (char*)d_ws;
    float4* packedR = (float4*)ws;                                         // 4 MB
    float4* packedS = (float4*)(ws + (size_t)UNITS * UNITS * 16);          // 1 MB
    float*  snum = (float*)(ws + (size_t)UNITS * UNITS * 16 + (size_t)SENS * UNITS * 16);
    float*  sden = snum + BATCH * UNITS;
    float*  stA  = sden + BATCH * UNITS;
    float*  stB  = stA + BATCH * UNITS;

    ltc_pack<<<UNITS * UNITS / 256, 256, 0, stream>>>(sigma, mu, w, erev, mask, packedR, UNITS * UNITS);
    ltc_pack<<<SENS * UNITS / 256, 256, 0, stream>>>(s_sigma, s_mu, s_w, s_erev, s_mask, packedS, SENS * UNITS);
    ltc_sensory<<<dim3(UNITS / 256, BATCH), 256, 0, stream>>>(inputs, in_w, in_b, packedS, snum, sden);

    float* outState = (float*)d_out + BATCH * MOTOR;
    const float* cur = state;
    for (int k = 0; k < UNFOLDS; ++k) {
        float* dst = (k == UNFOLDS - 1) ? outState : ((k & 1) ? stB : stA);
        ltc_unfold<<<dim3(UNITS / 256, BATCH / TB), 256, 0, stream>>>(
            cur, dst, packedR, gleak, vleak, cm, snum, sden);
        cur = dst;
    }
    ltc_output<<<(BATCH * MOTOR) / 256, 256, 0, stream>>>(outState, out_w, out_b, (float*)d_out);
}